// RetrieverModel_10926396801528
// MI455X (gfx1250) — compile-verified
//
#include <hip/hip_runtime.h>
#include <math.h>

// ---------------------------------------------------------------------------
// RetrieverModel pipeline for MI455X (gfx1250), fp32 end-to-end.
// Matrix ops use V_WMMA_F32_16X16X4_F32 (wave32, one wave per 16-wide N tile,
// all 4 M tiles fused so each data/label byte is read from HBM exactly once).
// ---------------------------------------------------------------------------

typedef float v2f __attribute__((ext_vector_type(2)));
typedef float v8f __attribute__((ext_vector_type(8)));

#define WMMA_F32(acc, a, b) \
  __builtin_amdgcn_wmma_f32_16x16x4_f32(false, (a), false, (b), (short)0, (acc), false, false)

// Problem constants (fixed by the reference)
#define BATCH 64
#define TIN   1500
#define TSEQ  45
#define FEAT  30
#define HID   256
#define DDIM  512
#define QDIM  (TSEQ * DDIM)   // 23040
#define NDATA 8000
#define NLAB  8000
#define KTOP  3
#define KCH   4               // K-chunks for the big NT GEMMs

// ---------------------------------------------------------------------------
// 1) x = (src perm/reshape) @ W1^T + b1    -> x[b][t][f], (64,45,30)
// ---------------------------------------------------------------------------
__global__ __launch_bounds__(256) void linear1_kernel(
    const float* __restrict__ src, const float* __restrict__ W1,
    const float* __restrict__ b1, float* __restrict__ x) {
  int id = blockIdx.x * 256 + threadIdx.x;           // b*1350 + t*30 + f
  if (id >= BATCH * TSEQ * FEAT) return;
  int f = id % FEAT;
  int t = (id / FEAT) % TSEQ;
  int b = id / (FEAT * TSEQ);
  const float* s = src + (size_t)b * TIN * FEAT + f; // src (B,1,1500,30)
  const float* w = W1 + (size_t)t * TIN;
  float acc = 0.f;
  for (int tt = 0; tt < TIN; ++tt) acc += s[(size_t)tt * FEAT] * w[tt];
  x[id] = acc + b1[t];
}

// ---------------------------------------------------------------------------
// 2) pre[dir][t][b][g] = x[b][t][:] @ Wih^T + (bih + bhh)   g in [0,1024)
// ---------------------------------------------------------------------------
__global__ __launch_bounds__(256) void pregate_kernel(
    const float* __restrict__ x,
    const float* __restrict__ Wih_f, const float* __restrict__ bih_f,
    const float* __restrict__ bhh_f,
    const float* __restrict__ Wih_b, const float* __restrict__ bih_b,
    const float* __restrict__ bhh_b, float* __restrict__ pre) {
  int id = blockIdx.x * 256 + threadIdx.x;   // exactly 2*45*64*1024 threads
  int g = id & 1023;
  int rest = id >> 10;
  int b = rest & 63;
  rest >>= 6;
  int t = rest % TSEQ;
  int dir = rest / TSEQ;
  const float* Wih = dir ? Wih_b : Wih_f;
  float acc = dir ? (bih_b[g] + bhh_b[g]) : (bih_f[g] + bhh_f[g]);
  const float* xr = x + (size_t)b * (TSEQ * FEAT) + (size_t)t * FEAT;
  const float* w = Wih + (size_t)g * FEAT;
#pragma unroll
  for (int f = 0; f < FEAT; ++f) acc += xr[f] * w[f];
  pre[id] = acc;   // layout dir*(45*64*1024) + t*(64*1024) + b*1024 + g == id
}

// ---------------------------------------------------------------------------
// 3) Persistent bidirectional LSTM: one workgroup per direction.
//    h in LDS (64KB), c in global scratch. Recurrent GEMM (64x256)@(256x1024)
//    via f32 WMMA; each wave owns i/f/g/o tiles of the same (b,j) range.
// ---------------------------------------------------------------------------
__global__ __launch_bounds__(1024) void lstm_fused(
    const float* __restrict__ pre_all, const float* __restrict__ Whh_f,
    const float* __restrict__ Whh_b, float* __restrict__ c_ws,
    float* __restrict__ q) {
  __shared__ float h_s[BATCH * HID];   // 64 KB
  const int dir = blockIdx.x;
  const float* pre = pre_all + (size_t)dir * TSEQ * BATCH * 1024;
  const float* Whh = dir ? Whh_b : Whh_f;
  float* c = c_ws + (size_t)dir * BATCH * HID;
  const int tid = threadIdx.x;
  for (int i = tid; i < BATCH * HID; i += 1024) { h_s[i] = 0.f; c[i] = 0.f; }
  __syncthreads();

  const int wave = tid >> 5;             // 0..31
  const int lane = tid & 31;
  const int lo = lane & 15;
  const int hi2 = (lane >> 4) << 1;      // 0 or 2 (K sub-offset)
  const int rbase = (lane >> 4) << 3;    // 0 or 8 (M sub-offset of C/D rows)

  for (int s = 0; s < TSEQ; ++s) {
    const int t = dir ? (TSEQ - 1 - s) : s;
    const float* pt = pre + (size_t)t * BATCH * 1024;
    v8f acc[2][4];
    // ---- recurrent GEMM: reads h_s ----
#pragma unroll
    for (int gi = 0; gi < 2; ++gi) {
      const int grp = wave * 2 + gi;           // 0..63
      const int m0 = (grp >> 4) << 4;          // batch tile
      const int j0 = (grp & 15) << 4;          // hidden-col tile
      const float* hrow = h_s + (m0 + lo) * HID + hi2;
#pragma unroll
      for (int gate = 0; gate < 4; ++gate) {
        const int n0 = gate * HID + j0;
        const float* wrow = Whh + (size_t)(n0 + lo) * HID + hi2;
        v8f a8 = {};
#pragma unroll 4
        for (int k = 0; k < HID; k += 4) {
          v2f a = *(const v2f*)(hrow + k);
          v2f b = *(const v2f*)(wrow + k);
          a8 = WMMA_F32(a8, a, b);
        }
        acc[gi][gate] = a8;
      }
    }
    __syncthreads();   // all waves done reading h_s
    // ---- gate nonlinearities + state update (register-local per wave) ----
#pragma unroll
    for (int gi = 0; gi < 2; ++gi) {
      const int grp = wave * 2 + gi;
      const int m0 = (grp >> 4) << 4;
      const int j0 = (grp & 15) << 4;
      const int j = j0 + lo;
#pragma unroll
      for (int r = 0; r < 8; ++r) {
        const int b = m0 + rbase + r;
        const float* pb = pt + (size_t)b * 1024;
        float iv = acc[gi][0][r] + pb[j];
        float fv = acc[gi][1][r] + pb[HID + j];
        float gv = acc[gi][2][r] + pb[2 * HID + j];
        float ov = acc[gi][3][r] + pb[3 * HID + j];
        float si = 1.f / (1.f + __expf(-iv));
        float sf = 1.f / (1.f + __expf(-fv));
        float so = 1.f / (1.f + __expf(-ov));
        float cc = sf * c[b * HID + j] + si * tanhf(gv);
        float hh = so * tanhf(cc);
        c[b * HID + j] = cc;
        h_s[b * HID + j] = hh;
        q[(size_t)b * QDIM + (size_t)t * DDIM + dir * HID + j] = hh;
      }
    }
    __syncthreads();
  }
}

// ---------------------------------------------------------------------------
// 4) Partial C(64xN) = A(64xK-slice) @ B(NxK-slice)^T, all 4 M-tiles fused in
//    one wave so each B row is streamed from HBM exactly once per K-slice.
//    Optionally accumulates per-row sum-of-squares of B (for cosine norms).
//    grid = (N/16, KCH); one wave per block.
// ---------------------------------------------------------------------------
__global__ __launch_bounds__(32) void gemm_nt_m64(
    const float* __restrict__ A, const float* __restrict__ Bm,
    float* __restrict__ Cpart, int N, int K, float* __restrict__ normpart) {
  const int n0 = blockIdx.x << 4;
  const int kc = blockIdx.y;
  const int klen = K / KCH;
  const int kbeg = kc * klen;
  const int lane = threadIdx.x & 31;
  const int lo = lane & 15;
  const int hi2 = (lane >> 4) << 1;
  const float* brow = Bm + (size_t)(n0 + lo) * K + kbeg + hi2;
  const float* abase = A + kbeg + hi2;
  v8f acc[4] = {};
  float ss = 0.f;
#pragma unroll 4
  for (int k = 0; k < klen; k += 4) {
    v2f b = *(const v2f*)(brow + k);
    ss += b.x * b.x + b.y * b.y;
#pragma unroll
    for (int mi = 0; mi < 4; ++mi) {
      v2f a = *(const v2f*)(abase + (size_t)((mi << 4) + lo) * K + k);
      acc[mi] = WMMA_F32(acc[mi], a, b);
    }
  }
  float* Cp = Cpart + (size_t)kc * BATCH * N;
  const int rbase = (lane >> 4) << 3;
#pragma unroll
  for (int mi = 0; mi < 4; ++mi)
#pragma unroll
    for (int r = 0; r < 8; ++r)
      Cp[(size_t)((mi << 4) + rbase + r) * N + (n0 + lo)] = acc[mi][r];
  if (normpart) {
    ss += __shfl_xor(ss, 16);   // lanes l and l+16 hold the two halves of row n0+l
    if (lane < 16) normpart[(size_t)kc * N + n0 + lo] = ss;
  }
}

// Reduce K-chunk partials: out[i] = alpha * sum_kc part[kc][i]
__global__ __launch_bounds__(256) void reduce_part(
    const float* __restrict__ part, float* __restrict__ out, int total,
    float alpha) {
  int i = blockIdx.x * 256 + threadIdx.x;
  if (i >= total) return;
  float s = 0.f;
#pragma unroll
  for (int c = 0; c < KCH; ++c) s += part[(size_t)c * total + i];
  out[i] = s * alpha;
}

// Finalize norms: out[i] = max(sqrt(sum_kc part[kc][i]), 1e-8)
__global__ __launch_bounds__(256) void norm_finalize(
    const float* __restrict__ part, float* __restrict__ out, int n) {
  int i = blockIdx.x * 256 + threadIdx.x;
  if (i >= n) return;
  float s = 0.f;
#pragma unroll
  for (int c = 0; c < KCH; ++c) s += part[(size_t)c * n + i];
  out[i] = fmaxf(sqrtf(s), 1e-8f);
}

// ---------------------------------------------------------------------------
// 5) C(64xN) = A(64xK) @ B(KxN), all 4 M-tiles fused per wave (B read once).
//    grid = N/16 blocks of one wave.
// ---------------------------------------------------------------------------
__global__ __launch_bounds__(32) void gemm_nn_m64(
    const float* __restrict__ A, const float* __restrict__ Bm,
    float* __restrict__ C, int N, int K) {
  const int n0 = blockIdx.x << 4;
  const int lane = threadIdx.x & 31;
  const int lo = lane & 15;
  const int hi2 = (lane >> 4) << 1;
  v8f acc[4] = {};
#pragma unroll 4
  for (int k = 0; k < K; k += 4) {
    v2f b;
    b.x = Bm[(size_t)(k + hi2) * N + n0 + lo];
    b.y = Bm[(size_t)(k + hi2 + 1) * N + n0 + lo];
#pragma unroll
    for (int mi = 0; mi < 4; ++mi) {
      v2f a = *(const v2f*)(A + (size_t)((mi << 4) + lo) * K + k + hi2);
      acc[mi] = WMMA_F32(acc[mi], a, b);
    }
  }
  const int rbase = (lane >> 4) << 3;
#pragma unroll
  for (int mi = 0; mi < 4; ++mi)
#pragma unroll
    for (int r = 0; r < 8; ++r)
      C[(size_t)((mi << 4) + rbase + r) * N + (n0 + lo)] = acc[mi][r];
}

// ---------------------------------------------------------------------------
// 6) Row softmax (in place), row L2 norms, top-3, gather
// ---------------------------------------------------------------------------
__global__ __launch_bounds__(256) void softmax_rows(float* __restrict__ p, int n) {
  __shared__ float red[256];
  const int tid = threadIdx.x;
  float* row = p + (size_t)blockIdx.x * n;
  float m = -3.402823466e38f;
  for (int i = tid; i < n; i += 256) m = fmaxf(m, row[i]);
  red[tid] = m; __syncthreads();
  for (int s = 128; s > 0; s >>= 1) { if (tid < s) red[tid] = fmaxf(red[tid], red[tid + s]); __syncthreads(); }
  m = red[0]; __syncthreads();
  float sum = 0.f;
  for (int i = tid; i < n; i += 256) { float e = __expf(row[i] - m); row[i] = e; sum += e; }
  red[tid] = sum; __syncthreads();
  for (int s = 128; s > 0; s >>= 1) { if (tid < s) red[tid] += red[tid + s]; __syncthreads(); }
  float inv = 1.f / red[0];
  for (int i = tid; i < n; i += 256) row[i] *= inv;
}

__global__ __launch_bounds__(256) void row_norms(
    const float* __restrict__ A, int n, float* __restrict__ out) {
  __shared__ float red[256];
  const int tid = threadIdx.x;
  const float* row = A + (size_t)blockIdx.x * n;
  float s = 0.f;
  for (int i = tid; i < n; i += 256) { float v = row[i]; s += v * v; }
  red[tid] = s; __syncthreads();
  for (int k = 128; k > 0; k >>= 1) { if (tid < k) red[tid] += red[tid + k]; __syncthreads(); }
  if (tid == 0) out[blockIdx.x] = fmaxf(sqrtf(red[0]), 1e-8f);
}

__global__ void topk3_kernel(const float* __restrict__ sim,
                             const float* __restrict__ ne,
                             const float* __restrict__ nl,
                             int* __restrict__ idx) {
  const int b = threadIdx.x;   // 64 threads, one row each
  const float* srow = sim + (size_t)b * NLAB;
  const float inv = 1.f / ne[b];
  float v0 = -3.402823466e38f, v1 = v0, v2 = v0;
  int i0 = 0, i1 = 0, i2 = 0;
  for (int l = 0; l < NLAB; ++l) {
    float v = srow[l] * inv / nl[l];
    if (v > v0)      { v2 = v1; i2 = i1; v1 = v0; i1 = i0; v0 = v; i0 = l; }
    else if (v > v1) { v2 = v1; i2 = i1; v1 = v;  i1 = l; }
    else if (v > v2) { v2 = v;  i2 = l; }
  }
  idx[b * KTOP + 0] = i0; idx[b * KTOP + 1] = i1; idx[b * KTOP + 2] = i2;
}

__global__ __launch_bounds__(256) void gather_labels(
    const float* __restrict__ label, const int* __restrict__ idx,
    float* __restrict__ out) {
  size_t i = (size_t)blockIdx.x * 256 + threadIdx.x;   // 64*3*23040 exact
  const int off = (int)(i % QDIM);
  const int bk = (int)(i / QDIM);
  const int k = bk % KTOP, b = bk / KTOP;
  out[(size_t)b * (KTOP * QDIM) + (size_t)k * QDIM + off] =
      label[(size_t)idx[b * KTOP + k] * QDIM + off];
}

// ---------------------------------------------------------------------------
extern "C" void kernel_launch(void* const* d_in, const int* in_sizes, int n_in,
                              void* d_out, int out_size, void* d_ws, size_t ws_size,
                              hipStream_t stream) {
  const float* src   = (const float*)d_in[0];
  const float* data  = (const float*)d_in[1];
  const float* label = (const float*)d_in[2];
  const float* W1    = (const float*)d_in[3];
  const float* b1    = (const float*)d_in[4];
  const float* Wih_f = (const float*)d_in[5];
  const float* Whh_f = (const float*)d_in[6];
  const float* bih_f = (const float*)d_in[7];
  const float* bhh_f = (const float*)d_in[8];
  const float* Wih_b = (const float*)d_in[9];
  const float* Whh_b = (const float*)d_in[10];
  const float* bih_b = (const float*)d_in[11];
  const float* bhh_b = (const float*)d_in[12];
  (void)in_sizes; (void)n_in; (void)out_size; (void)ws_size;

  // Workspace layout (floats): ~42 MB total
  float* ws     = (float*)d_ws;
  float* x      = ws;                       //     86,400
  float* pre    = x + 86400;                //  5,898,240  (fwd then bwd)
  float* c_ws   = pre + 5898240;            //     32,768
  float* q      = c_ws + 32768;             //  1,474,560  (B,45,512) flat
  float* part   = q + 1474560;              //  2,048,000  (KCH x 64 x 8000)
  float* npart  = part + 2048000;           //     32,000  (KCH x 8000)
  float* logits = npart + 32000;            //    512,000  (B,8000)
  float* sim    = logits + 512000;          //    512,000
  float* nlab   = sim + 512000;             //      8,000
  float* ne     = nlab + 8000;              //         64
  int*   idx    = (int*)(ne + 64);          //        192 ints

  float* e_imu  = (float*)d_out;            // 1,474,560 floats
  float* labout = e_imu + 1474560;          // 4,423,680 floats

  // 1) Linear(1500 -> 45)
  linear1_kernel<<<(BATCH * TSEQ * FEAT + 255) / 256, 256, 0, stream>>>(src, W1, b1, x);
  // 2) input-gate preactivations for both directions
  pregate_kernel<<<(2 * TSEQ * BATCH * 1024) / 256, 256, 0, stream>>>(
      x, Wih_f, bih_f, bhh_f, Wih_b, bih_b, bhh_b, pre);
  // 3) persistent bidirectional LSTM (one workgroup per direction)
  lstm_fused<<<2, 1024, 0, stream>>>(pre, Whh_f, Whh_b, c_ws, q);
  // 4) attention logits = q @ kv^T / sqrt(512): K-split partials, then reduce
  gemm_nt_m64<<<dim3(NDATA / 16, KCH), 32, 0, stream>>>(
      q, data, part, NDATA, QDIM, nullptr);
  reduce_part<<<(BATCH * NDATA + 255) / 256, 256, 0, stream>>>(
      part, logits, BATCH * NDATA, 0.04419417382415922f);
  // 5) softmax over N=8000
  softmax_rows<<<BATCH, 256, 0, stream>>>(logits, NDATA);
  // 6) e_imu = attn @ kv   (kv streamed exactly once)
  gemm_nn_m64<<<QDIM / 16, 32, 0, stream>>>(logits, data, e_imu, QDIM, NDATA);
  // 7) e_imu row norms (tiny)
  row_norms<<<BATCH, 256, 0, stream>>>(e_imu, QDIM, ne);
  // 8) sim = e_imu @ label^T with fused label sum-of-squares (label read once)
  gemm_nt_m64<<<dim3(NLAB / 16, KCH), 32, 0, stream>>>(
      e_imu, label, part, NLAB, QDIM, npart);
  reduce_part<<<(BATCH * NLAB + 255) / 256, 256, 0, stream>>>(
      part, sim, BATCH * NLAB, 1.0f);
  norm_finalize<<<(NLAB + 255) / 256, 256, 0, stream>>>(npart, nlab, NLAB);
  // 9) top-3 (with cosine scaling) and gather
  topk3_kernel<<<1, 64, 0, stream>>>(sim, ne, nlab, idx);
  gather_labels<<<(BATCH * KTOP * QDIM) / 256, 256, 0, stream>>>(label, idx, labout);
}